// DAREController_38628935860884
// MI455X (gfx1250) — compile-verified
//
#include <hip/hip_runtime.h>
#include <hip/hip_bf16.h>
#include <stdint.h>

#define BATCH 256
#define SEQ 32768
#define HALF (SEQ / 2)            /* elements copied by TDM; rest by async loads */
#define NTHREADS 512
#define NWAVES (NTHREADS / 32)    /* 16 */
#define KAPPA 64
#define RHO 0.5f

typedef uint32_t v4u __attribute__((ext_vector_type(4)));
typedef uint32_t v8u __attribute__((ext_vector_type(8)));
typedef float    v2f __attribute__((ext_vector_type(2)));
typedef float    v8f __attribute__((ext_vector_type(8)));

// Dynamic LDS layout (u32 words):
//   [0, SEQ)                       raw importance bits (TDM + async DMA from HBM)
//   [SEQ, +1024)                   eff bitmask (32768 bits)
//   [SEQ+1024, +4096)              per-wave histograms (16 x 256)
//   [SEQ+5120, +256)               merged histogram
//   [SEQ+5376, +16)                scalars: 0=nvalid 1=mmpre 2=k 3=prefix 4=kk 5=hardcnt 6=dropsum(f32)
#define OFF_EFF   (SEQ)
#define OFF_WHIST (SEQ + 1024)
#define OFF_HIST  (SEQ + 1024 + 4096)
#define OFF_U     (SEQ + 1024 + 4096 + 256)
#define SMEM_WORDS (OFF_U + 16)

__device__ __forceinline__ uint32_t fkey(uint32_t u) {
    // monotonic transform: larger float <-> larger unsigned key
    return u ^ ((u & 0x80000000u) ? 0xFFFFFFFFu : 0x80000000u);
}

extern "C" __global__ void dare_init(float* ws) {
    if (threadIdx.x < 4) ((uint32_t*)ws)[threadIdx.x] = 0u;
}

extern "C" __global__ __launch_bounds__(NTHREADS)
void dare_route(const float* __restrict__ imp, const int* __restrict__ mmask,
                float* __restrict__ out, float* __restrict__ ws) {
    extern __shared__ uint32_t smem[];
    uint32_t* s_raw   = smem;
    uint32_t* s_eff   = smem + OFF_EFF;
    uint32_t* s_whist = smem + OFF_WHIST;
    uint32_t* s_hist  = smem + OFF_HIST;
    uint32_t* s_u     = smem + OFF_U;

    const int row  = blockIdx.x;
    const int tid  = threadIdx.x;
    const int lane = tid & 31;
    const int wave = tid >> 5;

    const float* imp_row = imp + (size_t)row * SEQ;
    const int*   mm_row  = mmask + (size_t)row * SEQ;

    if (tid < 16) s_u[tid] = 0u;
    __syncthreads();

    const uint32_t lds_raw = (uint32_t)(uintptr_t)(void*)s_raw;  // LDS byte offset of s_raw
    const uint64_t gbase   = (uint64_t)(uintptr_t)imp_row;

    // ---- Phase 0a-1: wave 0 kicks a single TDM descriptor for the lower 64 KB ----
    if (wave == 0) {
        v4u g0;
        g0.x = 1u;                                            // count=1, user mode
        g0.y = lds_raw;                                       // lds_addr (bytes)
        g0.z = (uint32_t)gbase;                               // global_addr[31:0]
        g0.w = (uint32_t)((gbase >> 32) & 0x1FFFFFFu) | (2u << 30);  // addr[56:32], type=2
        v8u g1;
        g1.s0 = 2u << 16;                                     // data_size = 4 bytes
        g1.s1 = (uint32_t)(HALF & 0xFFFFu) << 16;             // tensor_dim0[15:0]
        g1.s2 = (uint32_t)(HALF >> 16) | (1u << 16);          // tensor_dim0[31:16], tensor_dim1=1
        g1.s3 = (uint32_t)(HALF & 0xFFFFu) << 16;             // tile_dim0 = HALF
        g1.s4 = 0u;                                           // tile_dim1/2 unused
        g1.s5 = (uint32_t)HALF;                               // tensor_dim0_stride
        g1.s6 = 0u;
        g1.s7 = 0u;
        asm volatile("tensor_load_to_lds %0, %1" :: "s"(g0), "s"(g1) : "memory");
    }

    // ---- Phase 0a-2: all waves async-DMA the upper 64 KB into LDS (ASYNCcnt path) ----
    #pragma unroll
    for (int i = 0; i < 8; ++i) {
        uint32_t boff = (uint32_t)(HALF * 4 + (i * NTHREADS + tid) * 16);
        uint32_t ldst = lds_raw + boff;
        asm volatile("global_load_async_to_lds_b128 %0, %1, %2"
                     :: "v"(ldst), "v"(boff), "s"(gbase) : "memory");
    }

    // ---- Phase 0b (overlapped with both DMAs): mask -> eff bitmask + counts ----
    {
        uint32_t nval_acc = 0, mmpre_acc = 0;
        for (int j = 0; j < 64; ++j) {
            int col = j * NTHREADS + tid;
            __builtin_prefetch(mm_row + col + 4 * NTHREADS, 0, 0);
            bool mm  = mm_row[col] != 0;
            bool eff = mm && (col >= KAPPA);
            uint32_t effb = (uint32_t)__ballot(eff);
            uint32_t mmb  = (uint32_t)__ballot(mm);
            if (lane == 0) {
                s_eff[j * NWAVES + wave] = effb;
                nval_acc += (uint32_t)__popc(effb);                        // n_valid
                if (j == 0 && wave < 2) mmpre_acc += (uint32_t)__popc(mmb); // mm in prefix
            }
        }
        if (lane == 0) {
            atomicAdd(&s_u[0], nval_acc);
            if (wave < 2) atomicAdd(&s_u[1], mmpre_acc);
        }
    }
    asm volatile("s_wait_asynccnt 0" ::: "memory");   // my async chunk landed
    __builtin_amdgcn_s_wait_tensorcnt(0);             // TDM chunk landed (no-op for waves 1..15)
    __syncthreads();

    if (tid == 0) {
        uint32_t nval = s_u[0], mmpre = s_u[1];
        float mmcnt  = (float)(nval + mmpre);
        float target = fmaxf(mmcnt * RHO, 1.0f);
        int k = min((int)target, (int)nval);
        s_u[2] = (uint32_t)k;
        s_u[3] = 0u;              // prefix
        s_u[4] = (uint32_t)k;     // running rank from the top
    }
    __syncthreads();
    const int k = (int)s_u[2];

    // ---- Phase 1: 4-pass MSB->LSB radix select with per-wave privatized histograms ----
    if (k > 0) {
        uint32_t* my_hist = s_whist + wave * 256;
        for (int pass = 0; pass < 4; ++pass) {
            const int shift = 24 - pass * 8;
            const uint32_t hmask = pass ? (0xFFFFFFFFu << (shift + 8)) : 0u;
            {   // zero all 16 private histograms (4096 words, b128 stores)
                v4u z = {0u, 0u, 0u, 0u};
                ((v4u*)s_whist)[tid * 2 + 0] = z;
                ((v4u*)s_whist)[tid * 2 + 1] = z;
            }
            __syncthreads();
            const uint32_t pref = s_u[3];
            for (int i = 0; i < 16; ++i) {
                int q = i * NTHREADS + tid;                    // quad index
                v4u v = ((const v4u*)s_raw)[q];
                uint32_t nib = (s_eff[q >> 3] >> ((q & 7) * 4)) & 0xFu;
                #pragma unroll
                for (int b = 0; b < 4; ++b) {
                    if ((nib >> b) & 1u) {
                        uint32_t key = fkey(v[b]);
                        if ((key & hmask) == pref)
                            atomicAdd(&my_hist[(key >> shift) & 255u], 1u);
                    }
                }
            }
            __syncthreads();

            // Merge the 16 private histograms with the matrix unit:
            // hist[j] = ones(1x16) . whist[0..15][j]  via 4 chained V_WMMA_F32_16X16X4_F32.
            // A = all-ones, so the column sum is invariant to the K-slot placement of the
            // 4 row values fed per step; only the lane<->N mapping matters (lanes 0-15 and
            // 16-31 both map to N=0..15 per the ISA layout). Wave w owns columns [16w,16w+16).
            {
                const int col  = wave * 16 + (lane & 15);
                const int half = lane >> 4;                    // 0 -> rows 4c+0/1, 1 -> rows 4c+2/3
                v2f a; a.x = 1.0f; a.y = 1.0f;
                v8f acc = {};
                #pragma unroll
                for (int c = 0; c < 4; ++c) {
                    const int r0 = 4 * c + 2 * half;
                    v2f bfrag;
                    bfrag.x = (float)s_whist[(r0 + 0) * 256 + col];
                    bfrag.y = (float)s_whist[(r0 + 1) * 256 + col];
                    acc = __builtin_amdgcn_wmma_f32_16x16x4_f32(
                        false, a, false, bfrag, (short)0, acc, false, false);
                }
                if (lane < 16) s_hist[wave * 16 + lane] = (uint32_t)acc[0];  // D row M=0
            }
            __syncthreads();

            if (wave == 0) {      // digit selection: block sums + wave32 suffix scan
                uint32_t kk = s_u[4];
                uint32_t s = 0;
                const int base_d = lane * 8;
                #pragma unroll
                for (int j = 0; j < 8; ++j) s += s_hist[base_d + j];
                uint32_t S = s;   // suffix sum over lanes (Hillis-Steele)
                #pragma unroll
                for (int off = 1; off < 32; off <<= 1) {
                    uint32_t t = __shfl_down(S, off, 32);
                    if (lane + off < 32) S += t;
                }
                uint32_t Snext = S - s;
                if ((S >= kk) && (Snext < kk)) {   // exactly one lane selected
                    uint32_t cum = Snext;
                    for (int j = 7; j >= 0; --j) {
                        uint32_t c = s_hist[base_d + j];
                        if (kk <= cum + c) {
                            s_u[3] |= ((uint32_t)(base_d + j)) << shift;
                            s_u[4]  = kk - cum;
                            break;
                        }
                        cum += c;
                    }
                }
            }
            __syncthreads();
        }
    }
    const uint32_t thresh = s_u[3];   // exact key of the k-th largest valid value

    // ---- Phase 2: emit keep mask (float4 stores) + per-thread reductions ----
    float drop_local = 0.0f;
    uint32_t hc = 0;
    float4* out_row4 = (float4*)(out + (size_t)row * SEQ);
    for (int i = 0; i < 16; ++i) {
        int q = i * NTHREADS + tid;
        v4u v = ((const v4u*)s_raw)[q];
        uint32_t nib = (s_eff[q >> 3] >> ((q & 7) * 4)) & 0xFu;
        bool pre = (q < (KAPPA / 4));
        float4 o;
        #pragma unroll
        for (int b = 0; b < 4; ++b) {
            bool ebit = (nib >> b) & 1u;
            bool hard = (k > 0) && ebit && (fkey(v[b]) >= thresh);
            ((float*)&o)[b] = (pre || hard) ? 1.0f : 0.0f;
            hc += hard ? 1u : 0u;
            if (ebit && !hard) drop_local += __uint_as_float(v[b]);
        }
        out_row4[q] = o;
    }
    atomicAdd(&s_u[5], hc);
    atomicAdd((float*)&s_u[6], drop_local);
    __syncthreads();

    if (tid == 0) {
        uint32_t nval = s_u[0], mmpre = s_u[1], hardcnt = s_u[5];
        float kept  = (float)(mmpre + hardcnt);
        float mmcnt = (float)(mmpre + nval);
        float ar = kept / (mmcnt + 1e-6f);
        atomicAdd(&ws[0], fabsf(ar - RHO));                 // sum |ar - rho|
        atomicAdd(&ws[1], fmaxf(RHO - ar, 0.0f));           // sum hinge
        atomicAdd(&ws[2], *((float*)&s_u[6]));              // dropped importance sum
        atomicAdd((unsigned int*)(ws + 3), nval - hardcnt); // dropped count
    }
}

extern "C" __global__ void dare_finalize(const float* ws, float* out) {
    if (threadIdx.x == 0 && blockIdx.x == 0) {
        float sum_abs = ws[0], sum_hinge = ws[1], drop_sum = ws[2];
        uint32_t ndrop = ((const uint32_t*)ws)[3];
        size_t base = (size_t)BATCH * SEQ;
        out[base + 0] = 1.0f * (sum_abs / (float)BATCH);                 // lambda_ratio * loss_ratio
        float dm = drop_sum / (float)(ndrop > 0u ? ndrop : 1u);
        out[base + 1] = 0.1f * (ndrop > 0u ? dm : 0.0f);                 // lambda_soft * loss_soft
        out[base + 2] = 1.0f * (sum_hinge / (float)BATCH);               // lambda_hard * loss_hard
    }
}

extern "C" void kernel_launch(void* const* d_in, const int* in_sizes, int n_in,
                              void* d_out, int out_size, void* d_ws, size_t ws_size,
                              hipStream_t stream) {
    (void)in_sizes; (void)n_in; (void)out_size; (void)ws_size;
    const float* imp = (const float*)d_in[0];
    const int*   mm  = (const int*)d_in[1];
    float* out = (float*)d_out;
    float* ws  = (float*)d_ws;

    dare_init<<<1, 32, 0, stream>>>(ws);
    size_t smem_bytes = (size_t)SMEM_WORDS * sizeof(uint32_t);  // ~149 KB of the 320 KB WGP LDS
    dare_route<<<BATCH, NTHREADS, smem_bytes, stream>>>(imp, mm, out, ws);
    dare_finalize<<<1, 32, 0, stream>>>(ws, out);
}